// PKSpellHierarchical_app2_1176821039634
// MI455X (gfx1250) — compile-verified
//
#include <hip/hip_runtime.h>

typedef float v2f __attribute__((ext_vector_type(2)));
typedef float v8f __attribute__((ext_vector_type(8)));
typedef unsigned long long ull;

#define T_LEN 1024
#define BATCH 128
#define DIN 17
#define H1 150
#define G1 450
#define H1P 160
#define G1P 464      /* LDS / gate width  (29 tiles) */
#define G1W 512      /* WMMA / packed-B width (32 tiles, 4 per wave) */
#define NT1 29       /* valid tiles */
#define NK1 40       /* H1P/4 */
#define H2 128
#define G2 384
#define NK2 32
#define LSEG 16
#define NSEG 64
#define NSEGTOT 8192   /* BATCH*NSEG */
#define HIERD 256
#define NPITCH 36
#define NKS 16
#define STK 556        /* 300 + 256 */

#define WMMA_F32(a, b, c) \
  __builtin_amdgcn_wmma_f32_16x16x4_f32(false, (a), false, (b), (short)0, (c), false, false)

#define V8F_ZERO ((v8f){0.f, 0.f, 0.f, 0.f, 0.f, 0.f, 0.f, 0.f})

__device__ __forceinline__ float sigm(float x) { return 1.f / (1.f + __expf(-x)); }

// ---------------------------------------------------------------- utility
__global__ void k_zero(float* p, ull n) {
  ull i = (ull)blockIdx.x * blockDim.x + threadIdx.x;
  ull st = (ull)gridDim.x * blockDim.x;
  for (; i < n; i += st) p[i] = 0.f;
}

// Pack main-GRU Whh into K-pair-interleaved B layout, padded to 160 x 512:
//   WhhP[dir][((k/2)*G1W + n)*2 + (k&1)] = Whh[n][k]   (zeros outside 150x450)
// WihP stays [dir][d][n<464] (scalar use, K=17).
__global__ void k_pack_main(const float* __restrict__ WhhF, const float* __restrict__ WhhB,
                            const float* __restrict__ WihF, const float* __restrict__ WihB,
                            float* __restrict__ WhhP, float* __restrict__ WihP) {
  int i = blockIdx.x * blockDim.x + threadIdx.x;
  int st = gridDim.x * blockDim.x;
  const int TOT1 = 2 * H1P * G1W;
  for (int j = i; j < TOT1; j += st) {
    int dir = j / (H1P * G1W), rem = j % (H1P * G1W);
    int k = rem / G1W, n = rem % G1W;
    const float* W = dir ? WhhB : WhhF;
    float v = (k < H1 && n < G1) ? W[n * H1 + k] : 0.f;
    int dst = dir * (H1P * G1W) + (((k >> 1) * G1W + n) << 1) + (k & 1);
    WhhP[dst] = v;
  }
  const int TOT2 = 2 * DIN * G1P;
  for (int j = i; j < TOT2; j += st) {
    int dir = j / (DIN * G1P), rem = j % (DIN * G1P);
    int d = rem / G1P, n = rem % G1P;
    const float* W = dir ? WihB : WihF;
    WihP[j] = (n < G1) ? W[n * DIN + d] : 0.f;
  }
}

__global__ void k_pack_hier(const float* __restrict__ WhhF, const float* __restrict__ WhhB,
                            const float* __restrict__ WihF, const float* __restrict__ WihB,
                            float* __restrict__ WhhP, float* __restrict__ WihP) {
  int i = blockIdx.x * blockDim.x + threadIdx.x;
  int st = gridDim.x * blockDim.x;
  const int TOT1 = 2 * H2 * G2;
  for (int j = i; j < TOT1; j += st) {
    int dir = j / (H2 * G2), rem = j % (H2 * G2);
    int k = rem / G2, n = rem % G2;
    const float* W = dir ? WhhB : WhhF;
    int dst = dir * (H2 * G2) + (((k >> 1) * G2 + n) << 1) + (k & 1);
    WhhP[dst] = W[n * H2 + k];
  }
  const int TOT2 = 2 * DIN * G2;
  for (int j = i; j < TOT2; j += st) {
    int dir = j / (DIN * G2), rem = j % (DIN * G2);
    int d = rem / G2, n = rem % G2;
    const float* W = dir ? WihB : WihF;
    WihP[j] = W[n * DIN + d];
  }
}

__global__ void k_pack_qv(const float* __restrict__ qW, const float* __restrict__ vW,
                          float* __restrict__ qWt, float* __restrict__ vWt) {
  int i = blockIdx.x * blockDim.x + threadIdx.x;
  int st = gridDim.x * blockDim.x;
  for (int j = i; j < HIERD * HIERD; j += st) {
    int k = j >> 8, o = j & 255;
    int dst = (((k >> 1) * HIERD + o) << 1) + (k & 1);
    qWt[dst] = qW[o * HIERD + k];
    vWt[dst] = vW[o * HIERD + k];
  }
}

// ---------------------------------------------------------------- main GRU step
// grid = 16 (2 dirs x 8 row-blocks of 16), 256 threads (8 waves).
// Uniform WMMA loop: every wave does exactly 4 tiles of the 512-wide padded
// gate matrix; only the 29 valid tiles are stored/used.
__global__ __launch_bounds__(256)
void k_main_step(const float* __restrict__ x, float* __restrict__ hBuf,
                 const float* __restrict__ WhhP, const float* __restrict__ WihP,
                 const float* __restrict__ bihF, const float* __restrict__ bhhF,
                 const float* __restrict__ bihB, const float* __restrict__ bhhB,
                 float* __restrict__ rnnOut, int t, int cur) {
  __shared__ float gh[16][G1P];
  __shared__ float xg[16][G1P];
  __shared__ float xs[16][DIN];
  const int dir = blockIdx.x >> 3;
  const int rb = blockIdx.x & 7;
  const int m0 = rb * 16;
  const int tid = threadIdx.x, lane = tid & 31, w = tid >> 5;
  const int teff = (dir == 0) ? t : (T_LEN - 1 - t);

  for (int i = tid; i < 16 * DIN; i += 256) {
    int m = i / DIN, d = i % DIN;
    xs[m][d] = x[((size_t)teff * BATCH + (m0 + m)) * DIN + d];
  }
  __syncthreads();

  const float* hprev = hBuf + ((size_t)dir * 2 + cur) * BATCH * H1P;
  float* hnext = hBuf + ((size_t)dir * 2 + (cur ^ 1)) * BATCH * H1P;
  const float* Bw = WhhP + (size_t)dir * H1P * G1W;  // pair-interleaved, 512 wide
  const float* Wx = WihP + (size_t)dir * DIN * G1P;
  const float* bih = dir ? bihB : bihF;
  const float* bhh = dir ? bhhB : bhhF;

  const int arow = m0 + (lane & 15);
  const int kofs = (lane >> 4) * 2;   // lanes 16-31 carry K+2,K+3
  const int nt0 = w * 4;
  const int lcol = lane & 15;

  v8f c[4] = {V8F_ZERO, V8F_ZERO, V8F_ZERO, V8F_ZERO};

  for (int kk = 0; kk < NK1; ++kk) {
    int kb = kk * 4 + kofs;                       // even
    v2f a = *(const v2f*)(hprev + (size_t)arow * H1P + kb);
    const float* bp = Bw + ((size_t)(kb >> 1) * G1W) * 2;
#pragma unroll
    for (int i = 0; i < 4; ++i) {
      int ncol = (nt0 + i) * 16 + lcol;
      v2f b = *(const v2f*)(bp + (ncol << 1));
      c[i] = WMMA_F32(a, b, c[i]);
    }
  }

#pragma unroll
  for (int i = 0; i < 4; ++i) {
    if (nt0 + i < NT1) {                          // store phase only
      int ncol = (nt0 + i) * 16 + lcol;
      float bihv = (ncol < G1) ? bih[ncol] : 0.f;
      float bhhv = (ncol < G1) ? bhh[ncol] : 0.f;
      float acc[8];
#pragma unroll
      for (int e = 0; e < 8; ++e) acc[e] = bihv;
      for (int d = 0; d < DIN; ++d) {
        float wv = Wx[(size_t)d * G1P + ncol];
#pragma unroll
        for (int e = 0; e < 8; ++e) {
          int m = (lane < 16) ? e : (8 + e);
          acc[e] += xs[m][d] * wv;
        }
      }
#pragma unroll
      for (int e = 0; e < 8; ++e) {
        int m = (lane < 16) ? e : (8 + e);
        gh[m][ncol] = c[i][e] + bhhv;
        xg[m][ncol] = acc[e];
      }
    }
  }
  __syncthreads();

  for (int i = tid; i < 16 * H1; i += 256) {
    int m = i / H1, j = i % H1;
    int gm = m0 + m;
    float r = sigm(xg[m][j] + gh[m][j]);
    float z = sigm(xg[m][H1 + j] + gh[m][H1 + j]);
    float nn = tanhf(xg[m][2 * H1 + j] + r * gh[m][2 * H1 + j]);
    float hp = hprev[(size_t)gm * H1P + j];
    float hn = (1.f - z) * nn + z * hp;
    hnext[(size_t)gm * H1P + j] = hn;
    rnnOut[((size_t)teff * BATCH + gm) * 300 + dir * H1 + j] = hn;
  }
}

// ---------------------------------------------------------------- hier GRU step
// grid = 1024 (2 dirs x 512 row-blocks), 256 threads. 24 tiles = 8 waves x 3.
__global__ __launch_bounds__(256)
void k_hier_step(const float* __restrict__ x, float* __restrict__ hBuf,
                 const float* __restrict__ WhhP, const float* __restrict__ WihP,
                 const float* __restrict__ bihF, const float* __restrict__ bhhF,
                 const float* __restrict__ bihB, const float* __restrict__ bhhB,
                 float* __restrict__ hierOut, int l, int cur) {
  __shared__ float gh[16][G2];
  __shared__ float xg[16][G2];
  __shared__ float xs[16][DIN];
  const int dir = blockIdx.x >> 9;
  const int rb = blockIdx.x & 511;
  const int m0 = rb * 16;
  const int tid = threadIdx.x, lane = tid & 31, w = tid >> 5;
  const int leff = (dir == 0) ? l : (LSEG - 1 - l);

  for (int i = tid; i < 16 * DIN; i += 256) {
    int m = i / DIN, d = i % DIN;
    int nseg = m0 + m;
    int b = nseg >> 6, s = nseg & 63;
    int tt = s * LSEG + leff;
    xs[m][d] = x[((size_t)tt * BATCH + b) * DIN + d];
  }
  __syncthreads();

  const float* hprev = hBuf + ((size_t)dir * 2 + cur) * NSEGTOT * H2;
  float* hnext = hBuf + ((size_t)dir * 2 + (cur ^ 1)) * NSEGTOT * H2;
  const float* Bw = WhhP + (size_t)dir * H2 * G2;  // pair-interleaved
  const float* Wx = WihP + (size_t)dir * DIN * G2;
  const float* bih = dir ? bihB : bihF;
  const float* bhh = dir ? bhhB : bhhF;

  const int arow = m0 + (lane & 15);
  const int kofs = (lane >> 4) * 2;
  const int nt0 = w * 3;
  const int lcol = lane & 15;

  v8f c[3] = {V8F_ZERO, V8F_ZERO, V8F_ZERO};

  for (int kk = 0; kk < NK2; ++kk) {
    int kb = kk * 4 + kofs;
    v2f a = *(const v2f*)(hprev + (size_t)arow * H2 + kb);
    const float* bp = Bw + ((size_t)(kb >> 1) * G2) * 2;
#pragma unroll
    for (int i = 0; i < 3; ++i) {
      int ncol = (nt0 + i) * 16 + lcol;
      v2f b = *(const v2f*)(bp + (ncol << 1));
      c[i] = WMMA_F32(a, b, c[i]);
    }
  }

#pragma unroll
  for (int i = 0; i < 3; ++i) {
    int ncol = (nt0 + i) * 16 + lcol;
    float bihv = bih[ncol];
    float bhhv = bhh[ncol];
    float acc[8];
#pragma unroll
    for (int e = 0; e < 8; ++e) acc[e] = bihv;
    for (int d = 0; d < DIN; ++d) {
      float wv = Wx[(size_t)d * G2 + ncol];
#pragma unroll
      for (int e = 0; e < 8; ++e) {
        int m = (lane < 16) ? e : (8 + e);
        acc[e] += xs[m][d] * wv;
      }
    }
#pragma unroll
    for (int e = 0; e < 8; ++e) {
      int m = (lane < 16) ? e : (8 + e);
      gh[m][ncol] = c[i][e] + bhhv;
      xg[m][ncol] = acc[e];
    }
  }
  __syncthreads();

  for (int i = tid; i < 16 * H2; i += 256) {
    int m = i / H2, j = i % H2;
    int gm = m0 + m;
    float r = sigm(xg[m][j] + gh[m][j]);
    float z = sigm(xg[m][H2 + j] + gh[m][H2 + j]);
    float nn = tanhf(xg[m][2 * H2 + j] + r * gh[m][2 * H2 + j]);
    float hp = hprev[(size_t)gm * H2 + j];
    float hn = (1.f - z) * nn + z * hp;
    hnext[(size_t)gm * H2 + j] = hn;
    hierOut[((size_t)leff * NSEGTOT + gm) * HIERD + dir * H2 + j] = hn;
  }
}

// ---------------------------------------------------------------- attention
__global__ __launch_bounds__(256)
void k_attention(const float* __restrict__ hierOut,
                 const float* __restrict__ qWt, const float* __restrict__ qb,
                 const float* __restrict__ vWt, const float* __restrict__ vb,
                 float* __restrict__ ctx) {
  __shared__ float vbuf[16][HIERD];
  __shared__ float qk[2][16][HIERD];
  __shared__ float sc[16][17];
  __shared__ float wsum[16];
  const int n = blockIdx.x;
  const int tid = threadIdx.x, lane = tid & 31, w = tid >> 5;

  for (int i = tid; i < 16 * HIERD; i += 256) {
    int l = i >> 8, d = i & 255;
    vbuf[l][d] = hierOut[((size_t)l * NSEGTOT + n) * HIERD + d];
  }
  __syncthreads();

  const int arow = lane & 15;
  const int kofs = (lane >> 4) * 2;
#pragma unroll
  for (int jj = 0; jj < 4; ++jj) {
    int job = w * 4 + jj;        // 0..15 -> Q tiles, 16..31 -> K tiles
    int which = job >> 4;
    int nt = job & 15;
    const float* Wt = which ? vWt : qWt;   // pair-interleaved
    const float* bb = which ? vb : qb;
    int ncol = nt * 16 + (lane & 15);
    v8f c = V8F_ZERO;
    for (int kk = 0; kk < HIERD / 4; ++kk) {
      int kb = kk * 4 + kofs;
      v2f a = *(const v2f*)(&vbuf[arow][kb]);
      v2f b = *(const v2f*)(Wt + (((size_t)(kb >> 1) * HIERD + ncol) << 1));
      c = WMMA_F32(a, b, c);
    }
    float bv = bb[ncol];
#pragma unroll
    for (int e = 0; e < 8; ++e) {
      int m = (lane < 16) ? e : (8 + e);
      qk[which][m][ncol] = c[e] + bv;
    }
  }
  __syncthreads();

  {
    int l = tid >> 4, m = tid & 15;
    float s = 0.f;
    for (int o = 0; o < HIERD; ++o) s += qk[0][l][o] * qk[1][m][o];
    sc[l][m] = s;
  }
  __syncthreads();

  if (tid < 16) {
    float mx = sc[tid][0];
    for (int m = 1; m < 16; ++m) mx = fmaxf(mx, sc[tid][m]);
    float sum = 0.f;
    for (int m = 0; m < 16; ++m) { float e = __expf(sc[tid][m] - mx); sc[tid][m] = e; sum += e; }
    float inv = 1.f / sum;
    for (int m = 0; m < 16; ++m) sc[tid][m] *= inv;
  }
  __syncthreads();
  if (tid < 16) {
    float s = 0.f;
    for (int l = 0; l < 16; ++l) s += sc[l][tid];
    wsum[tid] = s;
  }
  __syncthreads();
  {
    int d = tid;
    float s = 0.f;
    for (int m = 0; m < 16; ++m) s += wsum[m] * vbuf[m][d];
    ctx[(size_t)n * HIERD + d] = s;
  }
}

// ---------------------------------------------------------------- heads + loss
__global__ __launch_bounds__(256)
void k_head(const float* __restrict__ rnnOut, const float* __restrict__ ctx,
            const float* __restrict__ wpW, const float* __restrict__ wpB,
            const float* __restrict__ wkW, const float* __restrict__ wkB,
            const int* __restrict__ pitches, const int* __restrict__ keysigs,
            float* __restrict__ accum) {
  __shared__ float rowbuf[8][STK];
  __shared__ float plog[8][NPITCH];
  __shared__ float klog[8][NKS];
  __shared__ float red[8][4];
  const int row0 = blockIdx.x * 8;
  const int tid = threadIdx.x;

  for (int i = tid; i < 8 * STK; i += 256) {
    int rr = i / STK, k = i % STK;
    int row = row0 + rr;
    int t = row >> 7, b = row & 127;
    float v;
    if (k < 300) {
      v = rnnOut[(size_t)row * 300 + k];
    } else {
      int nseg = b * NSEG + (t >> 4);
      v = ctx[(size_t)nseg * HIERD + (k - 300)];
    }
    rowbuf[rr][k] = v;
  }
  __syncthreads();

  int g = tid >> 5, lane = tid & 31;
  for (int p = lane; p < NPITCH; p += 32) {
    float acc = wpB[p];
    for (int k = 0; k < STK; ++k) acc += rowbuf[g][k] * wpW[(size_t)p * STK + k];
    plog[g][p] = acc;
  }
  if (lane < NKS) {
    float acc = wkB[lane];
    for (int k = 0; k < HIERD; ++k) acc += rowbuf[g][300 + k] * wkW[(size_t)lane * HIERD + k];
    klog[g][lane] = acc;
  }
  __syncthreads();

  if (tid < 8) {
    int row = row0 + tid;
    int tp = pitches[row];
    float mx = plog[tid][0];
    for (int p = 1; p < NPITCH; ++p) mx = fmaxf(mx, plog[tid][p]);
    float s = 0.f;
    for (int p = 0; p < NPITCH; ++p) s += __expf(plog[tid][p] - mx);
    float lse = mx + __logf(s);
    float maskp = (tp != NPITCH - 1) ? 1.f : 0.f;
    float nllp = (lse - plog[tid][tp]) * maskp;

    int tk = keysigs[row];
    float mk = klog[tid][0];
    for (int p = 1; p < NKS; ++p) mk = fmaxf(mk, klog[tid][p]);
    float sk = 0.f;
    for (int p = 0; p < NKS; ++p) sk += __expf(klog[tid][p] - mk);
    float lsek = mk + __logf(sk);
    float maskk = (tk != NKS - 1) ? 1.f : 0.f;
    float nllk = (lsek - klog[tid][tk]) * maskk;

    red[tid][0] = nllp; red[tid][1] = maskp;
    red[tid][2] = nllk; red[tid][3] = maskk;
  }
  __syncthreads();
  if (tid == 0) {
    float a0 = 0.f, a1 = 0.f, a2 = 0.f, a3 = 0.f;
    for (int r = 0; r < 8; ++r) { a0 += red[r][0]; a1 += red[r][1]; a2 += red[r][2]; a3 += red[r][3]; }
    atomicAdd(&accum[0], a0);
    atomicAdd(&accum[1], a1);
    atomicAdd(&accum[2], a2);
    atomicAdd(&accum[3], a3);
  }
}

__global__ void k_finalize(const float* __restrict__ accum, float* __restrict__ out) {
  out[0] = accum[0] / fmaxf(accum[1], 1.f) + accum[2] / fmaxf(accum[3], 1.f);
}

// ---------------------------------------------------------------- launcher
extern "C" void kernel_launch(void* const* d_in, const int* in_sizes, int n_in,
                              void* d_out, int out_size, void* d_ws, size_t ws_size,
                              hipStream_t stream) {
  (void)in_sizes; (void)n_in; (void)out_size; (void)ws_size;
  const float* x = (const float*)d_in[0];
  const int* pitches = (const int*)d_in[1];
  const int* keysigs = (const int*)d_in[2];
  const float* rWihF = (const float*)d_in[5];
  const float* rWhhF = (const float*)d_in[6];
  const float* rbihF = (const float*)d_in[7];
  const float* rbhhF = (const float*)d_in[8];
  const float* rWihB = (const float*)d_in[9];
  const float* rWhhB = (const float*)d_in[10];
  const float* rbihB = (const float*)d_in[11];
  const float* rbhhB = (const float*)d_in[12];
  const float* hWihF = (const float*)d_in[13];
  const float* hWhhF = (const float*)d_in[14];
  const float* hbihF = (const float*)d_in[15];
  const float* hbhhF = (const float*)d_in[16];
  const float* hWihB = (const float*)d_in[17];
  const float* hWhhB = (const float*)d_in[18];
  const float* hbihB = (const float*)d_in[19];
  const float* hbhhB = (const float*)d_in[20];
  const float* qW = (const float*)d_in[21];
  const float* qb = (const float*)d_in[22];
  const float* vW = (const float*)d_in[23];
  const float* vb = (const float*)d_in[24];
  const float* wpW = (const float*)d_in[25];
  const float* wpB = (const float*)d_in[26];
  const float* wkW = (const float*)d_in[27];
  const float* wkB = (const float*)d_in[28];

  const size_t OFF_ACC   = 0;                 const size_t SZ_ACC   = 8;
  const size_t OFF_HMAIN = OFF_ACC + SZ_ACC;  const size_t SZ_HMAIN = 2ull * 2 * BATCH * H1P;
  const size_t OFF_HHIER = OFF_HMAIN + SZ_HMAIN; const size_t SZ_HHIER = 2ull * 2 * NSEGTOT * H2;
  const size_t OFF_WHHM  = OFF_HHIER + SZ_HHIER; const size_t SZ_WHHM = 2ull * H1P * G1W;
  const size_t OFF_WIHM  = OFF_WHHM + SZ_WHHM;   const size_t SZ_WIHM = 2ull * DIN * G1P;
  const size_t OFF_WHHH  = OFF_WIHM + SZ_WIHM;   const size_t SZ_WHHH = 2ull * H2 * G2;
  const size_t OFF_WIHH  = OFF_WHHH + SZ_WHHH;   const size_t SZ_WIHH = 2ull * DIN * G2;
  const size_t OFF_QWT   = OFF_WIHH + SZ_WIHH;   const size_t SZ_QWT  = (size_t)HIERD * HIERD;
  const size_t OFF_VWT   = OFF_QWT + SZ_QWT;     const size_t SZ_VWT  = (size_t)HIERD * HIERD;
  const size_t OFF_RNN   = OFF_VWT + SZ_VWT;     const size_t SZ_RNN  = (size_t)T_LEN * BATCH * 300;
  const size_t OFF_HOUT  = OFF_RNN + SZ_RNN;     const size_t SZ_HOUT = (size_t)LSEG * NSEGTOT * HIERD;
  const size_t OFF_CTX   = OFF_HOUT + SZ_HOUT;

  float* ws = (float*)d_ws;
  float* acc = ws + OFF_ACC;
  float* hMain = ws + OFF_HMAIN;
  float* hHier = ws + OFF_HHIER;
  float* WhhPm = ws + OFF_WHHM;
  float* WihPm = ws + OFF_WIHM;
  float* WhhPh = ws + OFF_WHHH;
  float* WihPh = ws + OFF_WIHH;
  float* qWt = ws + OFF_QWT;
  float* vWt = ws + OFF_VWT;
  float* rnnOut = ws + OFF_RNN;
  float* hierOut = ws + OFF_HOUT;
  float* ctx = ws + OFF_CTX;

  k_zero<<<2048, 256, 0, stream>>>(ws, (ull)(OFF_HHIER + SZ_HHIER));
  k_pack_main<<<1024, 256, 0, stream>>>(rWhhF, rWhhB, rWihF, rWihB, WhhPm, WihPm);
  k_pack_hier<<<512, 256, 0, stream>>>(hWhhF, hWhhB, hWihF, hWihB, WhhPh, WihPh);
  k_pack_qv<<<256, 256, 0, stream>>>(qW, vW, qWt, vWt);

  for (int l = 0; l < LSEG; ++l)
    k_hier_step<<<1024, 256, 0, stream>>>(x, hHier, WhhPh, WihPh,
                                          hbihF, hbhhF, hbihB, hbhhB,
                                          hierOut, l, l & 1);
  k_attention<<<NSEGTOT, 256, 0, stream>>>(hierOut, qWt, qb, vWt, vb, ctx);

  for (int t = 0; t < T_LEN; ++t)
    k_main_step<<<16, 256, 0, stream>>>(x, hMain, WhhPm, WihPm,
                                        rbihF, rbhhF, rbihB, rbhhB,
                                        rnnOut, t, t & 1);

  k_head<<<(T_LEN * BATCH) / 8, 256, 0, stream>>>(rnnOut, ctx, wpW, wpB, wkW, wkB,
                                                  pitches, keysigs, acc);
  k_finalize<<<1, 1, 0, stream>>>(acc, (float*)d_out);
}